// CBP_41970420417517
// MI455X (gfx1250) — compile-verified
//
#include <hip/hip_runtime.h>
#include <hip/hip_bf16.h>
#include <math.h>

// ---------------------------------------------------------------------------
// Compact Bilinear Pooling, reformulated exactly:
//   ifft(fft(p1)*fft(p2)).real == circular convolution of the two count
//   sketches. Summing over spatial positions first gives
//      y[b,d] = sum_{i,m} G_b[i,m] * s1[i]*s2[m] * [(h1[i]+h2[m]) % D == d]
//   with G_b = X_b X_b^T (512x512 Gram over 196 spatial positions).
// Work: 32 * 512*512*196 = 1.6 GMAC (WMMA f32) + 8.4M LDS scatter-adds.
// All inputs (12.8MB x + 32MB sketches) are L2-resident on MI455X (192MB L2).
// ---------------------------------------------------------------------------

typedef __attribute__((ext_vector_type(2))) float v2f;
typedef __attribute__((ext_vector_type(8))) float v8f;

#define CBP_C   512
#define CBP_D   8192
#define CBP_HW  196
#define CBP_B   32

// --- Kernel 1: recover (h, s) for each row of the dense count-sketch mats.
// One wave per row; exactly one nonzero (+-1) exists per row.
__global__ __launch_bounds__(256) void cbp_extract(
    const float* __restrict__ S1, const float* __restrict__ S2,
    int* __restrict__ h, float* __restrict__ s) {
  int gid  = blockIdx.x * 256 + threadIdx.x;
  int row  = gid >> 5;   // wave per row
  int lane = gid & 31;
  if (row >= 2 * CBP_C) return;
  const float* Srow = (row < CBP_C) ? (S1 + (size_t)row * CBP_D)
                                    : (S2 + (size_t)(row - CBP_C) * CBP_D);
  for (int j = lane; j < CBP_D; j += 32) {
    float v = Srow[j];
    if (v != 0.0f) { h[row] = j; s[row] = v; }   // unique writer per row
  }
}

// --- Kernel 2: zero the per-image accumulator (must be re-zeroed each call;
// the graph replays this launch sequence and the harness does not re-poison).
__global__ void cbp_zero(float* __restrict__ p, int n) {
  int i = blockIdx.x * blockDim.x + threadIdx.x;
  if (i < n) p[i] = 0.0f;
}

// --- Kernel 3: per-image Gram tiles via V_WMMA_F32_16X16X4_F32, scattered
// directly from the accumulator registers into an 8192-bin LDS histogram.
// Grid: (8, 32). Block: 256 threads = 8 waves. Each block owns 128 of the
// 1024 16x16 tiles of G_b and contributes a partial histogram.
__global__ __launch_bounds__(256) void cbp_gram_scatter(
    const float* __restrict__ x,      // [32][512][196]
    const int*   __restrict__ h,      // [1024] = h1 ++ h2
    const float* __restrict__ s,      // [1024] = s1 ++ s2
    float* __restrict__ yacc) {       // [32][8192]
  __shared__ float bins[CBP_D];       // 32 KB
  __shared__ int   hsh[2 * CBP_C];    // 4 KB
  __shared__ float ssh[2 * CBP_C];    // 4 KB

  const int tid = threadIdx.x;
  const int img = blockIdx.y;

  for (int i = tid; i < 2 * CBP_C; i += 256) { hsh[i] = h[i]; ssh[i] = s[i]; }
  for (int i = tid; i < CBP_D; i += 256)     bins[i] = 0.0f;
  __syncthreads();

  const float* Xb  = x + (size_t)img * CBP_C * CBP_HW;
  const int wave = tid >> 5;
  const int lane = tid & 31;
  const int lm   = lane & 15;        // M (A rows) / N (B cols) lane mapping
  const int kk   = (lane >> 4) * 2;  // half-wave K-slot split (consistent A/B)

  // Uniform-per-wave tile loop -> EXEC stays all-ones around WMMA.
  for (int tt = wave; tt < 128; tt += 8) {
    int tile = blockIdx.x * 128 + tt;        // 0..1023
    int i0 = (tile >> 5) * 16;               // G row block
    int m0 = (tile & 31) * 16;               // G col block
    const float* arow = Xb + (size_t)(i0 + lm) * CBP_HW;
    const float* brow = Xb + (size_t)(m0 + lm) * CBP_HW;

    v8f acc = {0.f, 0.f, 0.f, 0.f, 0.f, 0.f, 0.f, 0.f};
    for (int k0 = 0; k0 < CBP_HW; k0 += 4) {          // 196 = 49 * 4
      float2 av = *(const float2*)(arow + k0 + kk);   // 8B-aligned
      float2 bv = *(const float2*)(brow + k0 + kk);
      v2f a; a.x = av.x; a.y = av.y;
      v2f b; b.x = bv.x; b.y = bv.y;
      // 8 args: (neg_a, A, neg_b, B, c_mod, C, reuse_a, reuse_b)
      acc = __builtin_amdgcn_wmma_f32_16x16x4_f32(
          false, a, false, b, (short)0, acc, false, false);
    }

    // C/D layout: VGPR r -> M = r + 8*(lane>>4), N = lane&15.
    int   m    = m0 + lm;
    int   h2v  = hsh[CBP_C + m];
    float s2v  = ssh[CBP_C + m];
    int   mhi  = (lane >> 4) * 8;
#pragma unroll
    for (int r = 0; r < 8; ++r) {
      int   i   = i0 + r + mhi;
      float val = acc[r] * ssh[i] * s2v;
      int   bin = (hsh[i] + h2v) & (CBP_D - 1);
      atomicAdd(&bins[bin], val);               // ds_add_f32
    }
  }
  __syncthreads();

  float* yb = yacc + (size_t)img * CBP_D;
  for (int d = tid; d < CBP_D; d += 256)
    atomicAdd(&yb[d], bins[d]);                 // global_atomic_add_f32
}

// --- Kernel 4: signed sqrt + L2 normalize per image (one block per image).
__global__ __launch_bounds__(256) void cbp_finalize(
    const float* __restrict__ yacc, float* __restrict__ out) {
  __shared__ float red[256];
  const int b = blockIdx.x, tid = threadIdx.x;
  const float* yb = yacc + (size_t)b * CBP_D;

  float ss = 0.0f;
  for (int d = tid; d < CBP_D; d += 256) {
    float y = yb[d];
    float v = (y == 0.0f) ? 0.0f : copysignf(sqrtf(fabsf(y) + 1e-8f), y);
    ss += v * v;
  }
  red[tid] = ss;
  __syncthreads();
  for (int o = 128; o > 0; o >>= 1) {
    if (tid < o) red[tid] += red[tid + o];
    __syncthreads();
  }
  float inv = 1.0f / fmaxf(sqrtf(red[0]), 1e-12f);

  for (int d = tid; d < CBP_D; d += 256) {
    float y = yb[d];
    float v = (y == 0.0f) ? 0.0f : copysignf(sqrtf(fabsf(y) + 1e-8f), y);
    out[(size_t)b * CBP_D + d] = v * inv;
  }
}

extern "C" void kernel_launch(void* const* d_in, const int* in_sizes, int n_in,
                              void* d_out, int out_size, void* d_ws, size_t ws_size,
                              hipStream_t stream) {
  const float* x  = (const float*)d_in[0];   // [32,512,14,14] f32
  const float* S1 = (const float*)d_in[1];   // [512,8192] f32
  const float* S2 = (const float*)d_in[2];   // [512,8192] f32
  float* out = (float*)d_out;                // [32,8192] f32

  // Workspace layout (~1.03 MB total):
  char*  ws   = (char*)d_ws;
  int*   h    = (int*)ws;                    // 1024 ints   @ 0
  float* s    = (float*)(ws + 4096);         // 1024 floats @ 4KB
  float* yacc = (float*)(ws + 8192);         // 32*8192 f32 @ 8KB

  cbp_extract<<<dim3(128), dim3(256), 0, stream>>>(S1, S2, h, s);
  cbp_zero<<<dim3((CBP_B * CBP_D + 255) / 256), dim3(256), 0, stream>>>(
      yacc, CBP_B * CBP_D);
  cbp_gram_scatter<<<dim3(8, CBP_B), dim3(256), 0, stream>>>(x, h, s, yacc);
  cbp_finalize<<<dim3(CBP_B), dim3(256), 0, stream>>>(yacc, out);
}